// YoloLayer_77721728188987
// MI455X (gfx1250) — compile-verified
//
#include <hip/hip_runtime.h>
#include <hip/hip_bf16.h>
#include <stdint.h>

// ---------------- problem constants (from reference) ----------------
#define NB    128
#define NA    3
#define NCLS  20
#define NHW   52
#define NTGT  50
#define CHS   (NHW*NHW)        // 2704, stride between channels
#define BST   (75*CHS)         // 202800, stride between batches
#define RECF  16               // floats per target record
#define ROWSPC 13
#define NCHUNK 4
#define CELLS (ROWSPC*NHW)     // 676 cells per block
#define RECOFF 64              // byte offset of records in workspace

typedef float v2f __attribute__((ext_vector_type(2)));
typedef float v8f __attribute__((ext_vector_type(8)));
typedef int   v4i __attribute__((ext_vector_type(4)));

#if defined(__has_builtin)
# if __has_builtin(__builtin_amdgcn_global_load_async_to_lds_b128)
#  define HAVE_ASYNC_LDS 1
# endif
#endif

__device__ __forceinline__ float sigf(float x) { return 1.0f / (1.0f + __expf(-x)); }

__device__ __forceinline__ float bcef(float p, float t) {
    return -(t * __logf(fmaxf(p, 1e-38f)) + (1.0f - t) * __logf(fmaxf(1.0f - p, 1e-38f)));
}

// Wave32 sum via V_WMMA_F32_16X16X4_F32 with an all-ones B matrix.
// A layout (16x4 f32): lanes 0-15 -> A[lane][K=0] (vgpr0), lanes 16-31 -> A[lane-16][K=2].
// With B = ones: D[m][n] = v[m] + v[m+16] for every n, so summing this lane's 8
// C rows and adding the cross-half value (shfl_xor 16) yields the full 32-lane sum.
__device__ __forceinline__ float wave_sum_wmma(float v) {
    v2f a; a.x = v;    a.y = 0.0f;
    v2f b; b.x = 1.0f; b.y = 1.0f;
    v8f c = {0.f, 0.f, 0.f, 0.f, 0.f, 0.f, 0.f, 0.f};
    c = __builtin_amdgcn_wmma_f32_16x16x4_f32(false, a, false, b, (short)0, c, false, false);
    float s = c[0] + c[1] + c[2] + c[3] + c[4] + c[5] + c[6] + c[7];
    s += __shfl_xor(s, 16, 32);
    return s;
}

// ---------------- kernel 0: zero the 3 loss accumulators ----------------
__global__ void yolo_init(float* acc) {
    if (threadIdx.x < 3) acc[threadIdx.x] = 0.0f;
}

// ---------------- kernel 1: per-target prep (6400 items) ----------------
// record: [0]=l [1]=r [2]=t [3]=b [4]=gw [5]=gh [6]=area
//         [7]=best_anchor (-1 if invalid) [8]=gi [9]=gj
//         [10..13]=tvals(tx,ty,tw,th) [14]=t_iou [15]=cls_id
__global__ void yolo_prep(const float* __restrict__ out,
                          const float* __restrict__ tgt,
                          float* __restrict__ rec) {
    int idx = blockIdx.x * blockDim.x + threadIdx.x;
    if (idx >= NB * NTGT) return;
    int b = idx / NTGT;
    int t = idx - b * NTGT;
    const float* tb = tgt + (b * NTGT + t) * 5;
    float cls = tb[0], x = tb[1], y = tb[2], w = tb[3], h = tb[4];
    bool valid = (x != 0.0f);
    float gx = x * (float)NHW, gy = y * (float)NHW;
    float gw = w * 416.0f,     gh = h * 416.0f;

    // best anchor: centered-box IoU => inter = min(gw,aw)*min(gh,ah)
    int best = 0; float bestiou = -1.0f;
#pragma unroll
    for (int a = 0; a < NA; ++a) {
        float aw = (a == 0) ? 10.f : ((a == 1) ? 16.f : 33.f);
        float ah = (a == 0) ? 13.f : ((a == 1) ? 30.f : 23.f);
        float inter = fminf(gw, aw) * fminf(gh, ah);
        float uni   = gw * gh + aw * ah - inter;
        float iou   = inter / fmaxf(uni, 1e-12f);
        if (iou > bestiou) { bestiou = iou; best = a; }
    }
    float aw = (best == 0) ? 10.f : ((best == 1) ? 16.f : 33.f);
    float ah = (best == 0) ? 13.f : ((best == 1) ? 30.f : 23.f);
    int gi = (int)gx, gj = (int)gy;

    // pred box at the scatter cell (only 4 channel reads)
    int base = b * BST + best * 25 * CHS + gj * NHW + gi;
    float o0 = out[base], o1 = out[base + CHS];
    float o2 = out[base + 2 * CHS], o3 = out[base + 3 * CHS];
    float px = sigf(o0) + (float)gi;
    float py = sigf(o1) + (float)gj;
    float pw = __expf(o2) * aw, ph = __expf(o3) * ah;

    float uw = fmaxf(gx + gw * 0.5f, px + pw * 0.5f) - fminf(gx - gw * 0.5f, px - pw * 0.5f);
    float uh = fmaxf(gy + gh * 0.5f, py + ph * 0.5f) - fminf(gy - gh * 0.5f, py - ph * 0.5f);
    float cw = gw + pw - uw, ch2 = gh + ph - uh;
    float inter = (cw > 0.f && ch2 > 0.f) ? cw * ch2 : 0.f;
    float uni = gw * gh + pw * ph - inter;
    float tiou = inter / fmaxf(uni, 1e-12f);

    float* R = rec + idx * RECF;
    float hw = gw * 0.5f, hh = gh * 0.5f;
    // invalid targets naturally have gx=gy=gw=gh=0 -> degenerate box -> zero IoU
    R[0] = gx - hw; R[1] = gx + hw; R[2] = gy - hh; R[3] = gy + hh;
    R[4] = gw; R[5] = gh; R[6] = gw * gh;
    R[7] = valid ? (float)best : -1.0f;
    R[8] = (float)gi; R[9] = (float)gj;
    R[10] = gx - (float)gi; R[11] = gy - (float)gj;
    R[12] = __logf(fmaxf(gw, 1e-12f) / aw);
    R[13] = __logf(fmaxf(gh, 1e-12f) / ah);
    R[14] = tiou; R[15] = cls;
}

// ---------------- kernel 2: main per-cell pass ----------------
__global__ void __launch_bounds__(256)
yolo_main(const float* __restrict__ out, const float* __restrict__ rec,
          float* __restrict__ acc) {
    __shared__ float T[NTGT * RECF];   // 3200 B of staged target records
    __shared__ float red[8 * 3];
    const int tid = threadIdx.x;
    int blk   = blockIdx.x;
    int chunk = blk & 3;
    int a     = (blk >> 2) % NA;
    int b     = blk / (4 * NA);
    const float* rb = rec + b * NTGT * RECF;

    // stage the 50 records (3200 B) into LDS
#if defined(HAVE_ASYNC_LDS)
    if (tid < NTGT * RECF / 4) {
        const float* src = rb + tid * 4;
        float*       dst = &T[tid * 4];
        __builtin_amdgcn_global_load_async_to_lds_b128(
            (__attribute__((address_space(1))) v4i*)(uintptr_t)src,
            (__attribute__((address_space(3))) v4i*)(uint32_t)(uintptr_t)dst,
            0, 0);
    }
    asm volatile("s_wait_asynccnt 0x0" ::: "memory");
#else
    if (tid < NTGT * RECF / 4)
        reinterpret_cast<float4*>(T)[tid] = reinterpret_cast<const float4*>(rb)[tid];
#endif
    __syncthreads();

    float aw = (a == 0) ? 10.f : ((a == 1) ? 16.f : 33.f);
    float ah = (a == 0) ? 13.f : ((a == 1) ? 30.f : 23.f);
    float fa = (float)a;

    float s_coord = 0.f, s_conf = 0.f, s_cls = 0.f;
    for (int cell = tid; cell < CELLS; cell += 256) {
        int rrow = cell / NHW;
        int i = cell - rrow * NHW;
        int j = chunk * ROWSPC + rrow;
        int base = b * BST + a * 25 * CHS + j * NHW + i;
        float o0 = out[base],         o1 = out[base + CHS];
        float o2 = out[base + 2*CHS], o3 = out[base + 3*CHS];
        float o4 = out[base + 4*CHS];
        float sx = sigf(o0), sy = sigf(o1);
        float px = sx + (float)i, py = sy + (float)j;
        float pw = __expf(o2) * aw, ph = __expf(o3) * ah;
        float pl = px - pw * 0.5f, pr = px + pw * 0.5f;
        float pt = py - ph * 0.5f, pb = py + ph * 0.5f;
        float parea = pw * ph;
        float fi = (float)i, fj = (float)j;

        int bestt = -1; bool over = false;
        // division-free "max IoU > 0.5" test + scatter ownership (last target wins)
        for (int t = 0; t < NTGT; ++t) {
            const float* Tr = &T[t * RECF];
            float uw  = fmaxf(pr, Tr[1]) - fminf(pl, Tr[0]);
            float uh  = fmaxf(pb, Tr[3]) - fminf(pt, Tr[2]);
            float cw  = pw + Tr[4] - uw;
            float chh = ph + Tr[5] - uh;
            float inter = (cw > 0.f && chh > 0.f) ? cw * chh : 0.f;
            float uni   = parea + Tr[6] - inter;
            over = over || (inter > 0.5f * fmaxf(uni, 1e-12f));
            if (Tr[7] == fa && Tr[8] == fi && Tr[9] == fj) bestt = t;
        }

        float conf = sigf(o4);
        if (bestt >= 0) {                     // obj cell
            const float* Tr = &T[bestt * RECF];
            s_coord += bcef(sx, Tr[10]) + bcef(sy, Tr[11]);
            float dw = o2 - Tr[12], dh = o3 - Tr[13];
            s_coord += dw * dw + dh * dh;
            s_conf  += bcef(conf, Tr[14]);
            int ci = (int)Tr[15];
            const float* cb = out + base + 5 * CHS;
            float cs = 0.f;
#pragma unroll
            for (int c = 0; c < NCLS; ++c) {
                float xx = cb[c * CHS];
                cs += fmaxf(xx, 0.f) + __logf(1.f + __expf(-fabsf(xx)));
            }
            cs -= cb[ci * CHS];
            s_cls += cs;
        } else if (!over) {                   // noobj cell: bce(conf, 0)
            s_conf += -__logf(fmaxf(1.f - conf, 1e-38f));
        }                                     // masked cell: bce(0,0) == 0
    }

    // wave32 reduction through the matrix pipe (EXEC all-ones here)
    float r0 = wave_sum_wmma(s_coord);
    float r1 = wave_sum_wmma(s_conf);
    float r2 = wave_sum_wmma(s_cls);
    int wv = tid >> 5;
    if ((tid & 31) == 0) { red[wv*3+0] = r0; red[wv*3+1] = r1; red[wv*3+2] = r2; }
    __syncthreads();
    if (tid == 0) {
        float a0 = 0.f, a1 = 0.f, a2 = 0.f;
        for (int w = 0; w < 8; ++w) { a0 += red[w*3]; a1 += red[w*3+1]; a2 += red[w*3+2]; }
        atomicAdd(&acc[0], a0);
        atomicAdd(&acc[1], a1);
        atomicAdd(&acc[2], a2);
    }
}

// ---------------- kernel 3: combine with NaN guard ----------------
__global__ void yolo_final(const float* acc, float* out) {
    if (threadIdx.x == 0 && blockIdx.x == 0) {
        float lc  = acc[0] / (float)NB;
        float lf  = acc[1] / (float)NB;
        float lcl = acc[2] / (float)NB;
        float tot = lc + lf + lcl;
        out[0] = (lf != lf) ? (lc + lcl) : tot;
    }
}

extern "C" void kernel_launch(void* const* d_in, const int* in_sizes, int n_in,
                              void* d_out, int out_size, void* d_ws, size_t ws_size,
                              hipStream_t stream) {
    (void)in_sizes; (void)n_in; (void)out_size; (void)ws_size;
    const float* out = (const float*)d_in[0];
    const float* tgt = (const float*)d_in[1];
    float* acc = (float*)d_ws;
    float* rec = (float*)((char*)d_ws + RECOFF);

    yolo_init <<<1, 32, 0, stream>>>(acc);
    yolo_prep <<<(NB * NTGT + 255) / 256, 256, 0, stream>>>(out, tgt, rec);
    yolo_main <<<NB * NA * NCHUNK, 256, 0, stream>>>(out, rec, acc);
    yolo_final<<<1, 32, 0, stream>>>(acc, (float*)d_out);
}